// ReflectionFusion_81260781240715
// MI455X (gfx1250) — compile-verified
//
#include <hip/hip_runtime.h>

// ReflectionFusion on MI455X (gfx1250, wave32):
// out[b,c,h,w] = p*x + (1-p) * box9x9( sum_c x[b,c] ) / 81
// Separable 9-tap box sums expressed as banded-matrix products on
// V_WMMA_F32_16X16X4_F32 (f32 accuracy preserved; kernel is HBM-bound anyway).

typedef float v2f __attribute__((ext_vector_type(2)));
typedef float v8f __attribute__((ext_vector_type(8)));

#define BATCH 32
#define CHANS 3
#define HEIGHT 512
#define WIDTH 512
#define PAD 4
#define TILE 16
#define EXT 24 /* TILE + 2*PAD */

__global__ __launch_bounds__(32)
void ReflectionFusion_box9_wmma_kernel(const float* __restrict__ x,
                                       const float* __restrict__ p,
                                       float* __restrict__ out) {
    const int lane = threadIdx.x & 31;
    const int w0 = blockIdx.x * TILE;
    const int h0 = blockIdx.y * TILE;
    const int b  = blockIdx.z;

    __shared__ float s[EXT][EXT + 1]; // channel-sum, 24x24 halo region
    __shared__ float t[EXT][TILE];    // horizontally box-summed, 24x16

    const size_t planeHW = (size_t)HEIGHT * WIDTH;
    const float* xb = x + (size_t)b * CHANS * planeHW;
    const float* pb = p + (size_t)b * planeHW;
    float* ob = out + (size_t)b * CHANS * planeHW;

    // ---- Stage 1: channel-sum of the 24x24 halo region into LDS (zero pad) ----
    for (int idx = lane; idx < EXT * EXT; idx += 32) {
        const int r = idx / EXT;
        const int c = idx - r * EXT;
        const int gr = h0 - PAD + r;
        const int gc = w0 - PAD + c;
        float v = 0.0f;
        if (gr >= 0 && gr < HEIGHT && gc >= 0 && gc < WIDTH) {
            const size_t off = (size_t)gr * WIDTH + gc;
            v = xb[off] + xb[planeHW + off] + xb[2 * planeHW + off];
        }
        s[r][c] = v;
    }
    __syncthreads();

    // WMMA f32 lane mapping (ISA 7.12.2, wave32):
    //  A 16x4 : lane L holds M = L&15; VGPR v holds K = 2*(L>>4) + v
    //  B 4x16 : lane L holds N = L&15; VGPR v holds K = 2*(L>>4) + v
    //  C/D    : lane L holds N = L&15; VGPR r holds M = r + 8*(L>>4)
    const int half = lane >> 4;  // 0 | 1
    const int mrow = lane & 15;  // M for A, N for B/C/D

    // ---- Stage 2: horizontal 9-tap box sum, t(24x16) = s x band ----
    // t[r][j] = sum_{c=j..j+8} s[r][c];  band B_h[c][j] = (j <= c <= j+8)
    #pragma unroll
    for (int rb = 0; rb < 2; ++rb) {
        const int rowbase = rb * 8; // rows 0-15, then 8-23 (overlap rows equal)
        v8f acc = {};
        #pragma unroll
        for (int k = 0; k < 6; ++k) {
            const int k0 = 4 * k + 2 * half;
            v2f a, bm;
            a.x = s[rowbase + mrow][k0];
            a.y = s[rowbase + mrow][k0 + 1];
            bm.x = (mrow <= k0     && k0     <= mrow + 8) ? 1.0f : 0.0f;
            bm.y = (mrow <= k0 + 1 && k0 + 1 <= mrow + 8) ? 1.0f : 0.0f;
            acc = __builtin_amdgcn_wmma_f32_16x16x4_f32(
                false, a, false, bm, (short)0, acc, false, false);
        }
        #pragma unroll
        for (int r = 0; r < 8; ++r)
            t[rowbase + r + 8 * half][mrow] = acc[r];
    }
    __syncthreads();

    // ---- Stage 3: vertical 9-tap box sum, D(16x16) = band x t(24x16) ----
    // D[h][j] = sum_{r=h..h+8} t[r][j];  band A_v[h][r] = (h <= r <= h+8)
    v8f acc = {};
    #pragma unroll
    for (int k = 0; k < 6; ++k) {
        const int k0 = 4 * k + 2 * half;
        v2f a, bm;
        a.x = (mrow <= k0     && k0     <= mrow + 8) ? 1.0f : 0.0f;
        a.y = (mrow <= k0 + 1 && k0 + 1 <= mrow + 8) ? 1.0f : 0.0f;
        bm.x = t[k0][mrow];
        bm.y = t[k0 + 1][mrow];
        acc = __builtin_amdgcn_wmma_f32_16x16x4_f32(
            false, a, false, bm, (short)0, acc, false, false);
    }

    // ---- Stage 4: blend out = p*x + (1-p)*boxsum/81, all 3 channels ----
    const float inv81 = 1.0f / 81.0f;
    #pragma unroll
    for (int r = 0; r < 8; ++r) {
        const int gh = h0 + r + 8 * half;
        const int gw = w0 + mrow;
        const size_t off = (size_t)gh * WIDTH + gw;
        const float m = acc[r] * inv81;
        const float pv = pb[off];
        const float q = 1.0f - pv;
        #pragma unroll
        for (int c = 0; c < CHANS; ++c) {
            const size_t co = c * planeHW + off;
            ob[co] = pv * xb[co] + q * m;
        }
    }
}

extern "C" void kernel_launch(void* const* d_in, const int* in_sizes, int n_in,
                              void* d_out, int out_size, void* d_ws, size_t ws_size,
                              hipStream_t stream) {
    (void)in_sizes; (void)n_in; (void)out_size; (void)d_ws; (void)ws_size;
    const float* x = (const float*)d_in[0];
    const float* p = (const float*)d_in[1];
    float* out = (float*)d_out;

    dim3 grid(WIDTH / TILE, HEIGHT / TILE, BATCH); // 32 x 32 x 32 tiles
    dim3 block(32);                                // one wave32 per 16x16 tile
    hipLaunchKernelGGL(ReflectionFusion_box9_wmma_kernel, grid, block, 0, stream,
                       x, p, out);
}